// DBN_Single_17987323036451
// MI455X (gfx1250) — compile-verified
//
#include <hip/hip_runtime.h>

typedef __attribute__((ext_vector_type(2))) float v2f;
typedef __attribute__((ext_vector_type(4))) float v4f;
typedef __attribute__((ext_vector_type(8))) float v8f;
typedef __attribute__((ext_vector_type(4))) int   v4i;

#define CCH    64
#define HW     3136                 // 56*56
#define NBATCH 128
#define MTOT   (NBATCH * HW)        // 401408 samples per channel
#define CHW    (CCH * HW)
#define KC     64                   // k-chunk size; 3136/64 = 49 chunks per image
#define NCHUNK (MTOT / KC)          // 6272
#define ROWP   68                   // padded LDS row stride: 272B -> 16B-aligned rows,
                                    // bank stride 4 -> conflict-free 16-lane fragments
#define EPSV   1e-3f
#define NITER  18

#if __has_builtin(__builtin_amdgcn_global_load_async_to_lds_b128)
#define HAS_ASYNC 1
#else
#define HAS_ASYNC 0
#endif

typedef __attribute__((address_space(1))) v4i gas_v4i;   // global int4
typedef __attribute__((address_space(3))) v4i las_v4i;   // LDS int4

// 16-byte HBM -> LDS copy: async (ASYNCcnt-tracked, no VGPR round trip) if available
__device__ __forceinline__ void lds_copy16(const float* g, float* l) {
#if HAS_ASYNC
    __builtin_amdgcn_global_load_async_to_lds_b128(
        (gas_v4i*)(const_cast<float*>(g)),
        (las_v4i*)l, 0, 0);
#else
    *(v4f*)l = *(const v4f*)g;
#endif
}

template <int N>
__device__ __forceinline__ void async_wait() {
#if HAS_ASYNC
#if __has_builtin(__builtin_amdgcn_s_wait_asynccnt)
    __builtin_amdgcn_s_wait_asynccnt(N);
#else
    asm volatile("s_wait_asynccnt %0" ::"n"(N) : "memory");
#endif
#endif
}

// ---------------------------------------------------------------------------
// zero workspace accumulators (sums + Gram)
// ---------------------------------------------------------------------------
__global__ __launch_bounds__(512) void dbn_zero_kernel(float* p, int n) {
    int i = blockIdx.x * blockDim.x + threadIdx.x;
    if (i < n) p[i] = 0.0f;
}

// ---------------------------------------------------------------------------
// Pass 1: per-channel sums + raw Gram G = X * X^T (64x64), WMMA f32,
// double-buffered async tile loads.
// ---------------------------------------------------------------------------
__global__ __launch_bounds__(512) void dbn_gram_kernel(const float* __restrict__ X,
                                                       float* __restrict__ sums,
                                                       float* __restrict__ G) {
    __shared__ float Xs[2][CCH * ROWP];   // [buf][channel][k]

    const int tid   = threadIdx.x;
    const int lane  = tid & 31;
    const int w     = tid >> 5;          // wave 0..15
    const int c1    = (w >> 2) * 16;     // A-tile channel base (M)
    const int c2    = (w & 3) * 16;      // B-tile channel base (N)
    const int mrow  = lane & 15;
    const int khalf = lane >> 4;

    const int lc = tid >> 3;             // channel this thread loads
    const int lj = (tid & 7) * 8;        // k offset within chunk (32B aligned)

    v8f   acc  = {0.f, 0.f, 0.f, 0.f, 0.f, 0.f, 0.f, 0.f};
    float sacc = 0.f;

    auto issue = [&](int chunk, int buf) {
        const int n   = chunk / 49;
        const int hw0 = (chunk - n * 49) * KC;
        const float* src = X + (size_t)n * CHW + (size_t)lc * HW + hw0 + lj;
        float* dst = &Xs[buf][lc * ROWP + lj];
        lds_copy16(src, dst);
        lds_copy16(src + 4, dst + 4);
    };

    int cur   = 0;
    int chunk = blockIdx.x;
    if (chunk < NCHUNK) issue(chunk, 0);

    for (; chunk < NCHUNK; chunk += gridDim.x) {
        const int nxt = chunk + gridDim.x;
        if (nxt < NCHUNK) {               // uniform branch per block
            issue(nxt, cur ^ 1);
            async_wait<2>();              // current buffer resident (in-order)
        } else {
            async_wait<0>();
        }
        __syncthreads();

        const float* T = Xs[cur];

        // fold per-channel sum (reads this thread's own 8 values)
        const float* mine = &T[lc * ROWP + lj];
        v4f m0 = *(const v4f*)mine;
        v4f m1 = *(const v4f*)(mine + 4);
        sacc += m0.x + m0.y + m0.z + m0.w + m1.x + m1.y + m1.z + m1.w;

        #pragma unroll
        for (int k0 = 0; k0 < KC; k0 += 4) {
            const int kk = k0 + khalf * 2;
            v2f a = *(const v2f*)&T[(c1 + mrow) * ROWP + kk];   // A[M,K] = x[c1+M, K]
            v2f b = *(const v2f*)&T[(c2 + mrow) * ROWP + kk];   // B[K,N] = x[c2+N, K]
            acc = __builtin_amdgcn_wmma_f32_16x16x4_f32(false, a, false, b,
                                                        (short)0, acc, false, false);
        }
        __syncthreads();                  // all reads of buf[cur] done
        cur ^= 1;
    }

    #pragma unroll
    for (int r = 0; r < 8; ++r) {
        const int gr = c1 + r + khalf * 8;
        const int gc = c2 + mrow;
        atomicAdd(&G[gr * CCH + gc], acc[r]);
    }
    atomicAdd(&sums[lc], sacc);
}

// ---------------------------------------------------------------------------
// 64x64 matmul tile for one wave, operands in padded LDS, WMMA f32
// ---------------------------------------------------------------------------
__device__ inline v8f dbn_mm_tile(const float* A, const float* B,
                                  int r0, int c0, int mrow, int khalf) {
    v8f acc = {0.f, 0.f, 0.f, 0.f, 0.f, 0.f, 0.f, 0.f};
    #pragma unroll
    for (int k0 = 0; k0 < 64; k0 += 4) {
        const int kk = k0 + khalf * 2;
        v2f a = *(const v2f*)&A[(r0 + mrow) * ROWP + kk];
        v2f b;
        b.x = B[kk * ROWP + c0 + mrow];
        b.y = B[(kk + 1) * ROWP + c0 + mrow];
        acc = __builtin_amdgcn_wmma_f32_16x16x4_f32(false, a, false, b,
                                                    (short)0, acc, false, false);
    }
    return acc;
}

// ---------------------------------------------------------------------------
// Pass 2 (single block): sigma -> Newton-Schulz inverse sqrt -> WM, bias
// sigma = (G - s s^T/m)/m + eps*I ; WM = sigma^{-1/2} ; bias = WM @ mean
// ---------------------------------------------------------------------------
__global__ __launch_bounds__(512) void dbn_invsqrt_kernel(const float* __restrict__ sums,
                                                          const float* __restrict__ G,
                                                          float* __restrict__ WM,
                                                          float* __restrict__ BIAS) {
    __shared__ float Y[CCH * ROWP];
    __shared__ float Z[CCH * ROWP];
    __shared__ float P[CCH * ROWP];
    __shared__ float sdiag[CCH];
    __shared__ float strace;

    const int tid   = threadIdx.x;
    const int lane  = tid & 31;
    const int w     = tid >> 5;
    const int r0    = (w >> 2) * 16;
    const int c0    = (w & 3) * 16;
    const int mrow  = lane & 15;
    const int khalf = lane >> 4;
    const float inv_m = 1.0f / (float)MTOT;

    for (int idx = tid; idx < CCH * CCH; idx += 512) {
        const int i = idx >> 6, j = idx & 63;
        float v = (G[idx] - sums[i] * sums[j] * inv_m) * inv_m + (i == j ? EPSV : 0.f);
        Y[i * ROWP + j] = v;
        Z[i * ROWP + j] = (i == j) ? 1.f : 0.f;
    }
    __syncthreads();
    if (tid < CCH) sdiag[tid] = Y[tid * ROWP + tid];
    __syncthreads();
    if (tid == 0) {
        float t = 0.f;
        for (int i = 0; i < CCH; ++i) t += sdiag[i];
        strace = t;
    }
    __syncthreads();
    const float tr   = strace;
    const float invt = 1.0f / tr;
    for (int idx = tid; idx < CCH * CCH; idx += 512) {
        const int i = idx >> 6, j = idx & 63;
        Y[i * ROWP + j] *= invt;                      // Y0 = sigma/tr, eigs in (0,1]
    }
    __syncthreads();

    for (int it = 0; it < NITER; ++it) {
        v8f p = dbn_mm_tile(Z, Y, r0, c0, mrow, khalf);   // P = Z@Y
        #pragma unroll
        for (int r = 0; r < 8; ++r)
            P[(r0 + r + khalf * 8) * ROWP + c0 + mrow] = p[r];
        __syncthreads();
        for (int idx = tid; idx < CCH * CCH; idx += 512) { // P = 0.5*(3I - P)
            const int i = idx >> 6, j = idx & 63;
            float v = P[i * ROWP + j];
            P[i * ROWP + j] = 0.5f * ((i == j ? 3.f : 0.f) - v);
        }
        __syncthreads();
        v8f ny = dbn_mm_tile(Y, P, r0, c0, mrow, khalf);   // Y <- Y@P
        v8f nz = dbn_mm_tile(P, Z, r0, c0, mrow, khalf);   // Z <- P@Z
        __syncthreads();
        #pragma unroll
        for (int r = 0; r < 8; ++r) {
            const int row = (r0 + r + khalf * 8) * ROWP + c0 + mrow;
            Y[row] = ny[r];
            Z[row] = nz[r];
        }
        __syncthreads();
    }

    const float scale = rsqrtf(tr);                   // sigma^{-1/2} = Z/sqrt(tr)
    for (int idx = tid; idx < CCH * CCH; idx += 512) {
        const int i = idx >> 6, j = idx & 63;
        WM[idx] = Z[i * ROWP + j] * scale;
    }
    __syncthreads();
    if (tid < CCH) {
        float b = 0.f;
        for (int j = 0; j < CCH; ++j)
            b += Z[tid * ROWP + j] * scale * (sums[j] * inv_m);
        BIAS[tid] = b;
    }
}

// ---------------------------------------------------------------------------
// Pass 3: out = WM @ x - bias (64 x m), WMMA f32, double-buffered async tiles
// ---------------------------------------------------------------------------
__global__ __launch_bounds__(512) void dbn_apply_kernel(const float* __restrict__ X,
                                                        const float* __restrict__ WM,
                                                        const float* __restrict__ BIAS,
                                                        float* __restrict__ out) {
    __shared__ float Xs[2][CCH * ROWP];
    __shared__ float Ws[CCH * ROWP];
    __shared__ float Bs[CCH];

    const int tid   = threadIdx.x;
    const int lane  = tid & 31;
    const int w     = tid >> 5;
    const int r0    = (w >> 2) * 16;   // output-channel tile (M)
    const int c0    = (w & 3) * 16;    // k tile (N)
    const int mrow  = lane & 15;
    const int khalf = lane >> 4;

    for (int idx = tid; idx < CCH * CCH; idx += 512) {
        const int i = idx >> 6, j = idx & 63;
        Ws[i * ROWP + j] = WM[idx];
    }
    if (tid < CCH) Bs[tid] = BIAS[tid];

    const int lc = tid >> 3;
    const int lj = (tid & 7) * 8;

    auto issue = [&](int chunk, int buf) {
        const int n   = chunk / 49;
        const int hw0 = (chunk - n * 49) * KC;
        const float* src = X + (size_t)n * CHW + (size_t)lc * HW + hw0 + lj;
        float* dst = &Xs[buf][lc * ROWP + lj];
        lds_copy16(src, dst);
        lds_copy16(src + 4, dst + 4);
    };

    int cur   = 0;
    int chunk = blockIdx.x;
    if (chunk < NCHUNK) issue(chunk, 0);

    for (; chunk < NCHUNK; chunk += gridDim.x) {
        const int nxt = chunk + gridDim.x;
        if (nxt < NCHUNK) {
            issue(nxt, cur ^ 1);
            async_wait<2>();
        } else {
            async_wait<0>();
        }
        __syncthreads();                 // tile (and Ws/Bs on 1st iter) visible

        const float* T = Xs[cur];
        v8f acc = {0.f, 0.f, 0.f, 0.f, 0.f, 0.f, 0.f, 0.f};
        #pragma unroll
        for (int k0 = 0; k0 < 64; k0 += 4) {   // K dim = input channel c'
            const int kk = k0 + khalf * 2;
            v2f a = *(const v2f*)&Ws[(r0 + mrow) * ROWP + kk];  // A[M,K]=WM[r0+M,c']
            v2f b;
            b.x = T[kk * ROWP + c0 + mrow];                     // B[K,N]=x[c',kbase+N]
            b.y = T[(kk + 1) * ROWP + c0 + mrow];
            acc = __builtin_amdgcn_wmma_f32_16x16x4_f32(false, a, false, b,
                                                        (short)0, acc, false, false);
        }

        const int n   = chunk / 49;
        const int hw  = (chunk - n * 49) * KC + c0 + mrow;
        float* obase  = out + (size_t)n * CHW + hw;
        #pragma unroll
        for (int r = 0; r < 8; ++r) {
            const int ch = r0 + r + khalf * 8;
            obase[(size_t)ch * HW] = acc[r] - Bs[ch];
        }
        __syncthreads();                 // all reads of buf[cur] done
        cur ^= 1;
    }
}

// ---------------------------------------------------------------------------
extern "C" void kernel_launch(void* const* d_in, const int* in_sizes, int n_in,
                              void* d_out, int out_size, void* d_ws, size_t ws_size,
                              hipStream_t stream) {
    const float* X = (const float*)d_in[0];
    float* out = (float*)d_out;

    float* sums = (float*)d_ws;          // 64
    float* G    = sums + 64;             // 4096
    float* WM   = G + 4096;              // 4096
    float* BIAS = WM + 4096;             // 64

    dbn_zero_kernel<<<9, 512, 0, stream>>>(sums, 64 + 4096);
    dbn_gram_kernel<<<256, 512, 0, stream>>>(X, sums, G);
    dbn_invsqrt_kernel<<<1, 512, 0, stream>>>(sums, G, WM, BIAS);
    dbn_apply_kernel<<<512, 512, 0, stream>>>(X, WM, BIAS, out);
}